// GPT2Attention_69295002353747
// MI455X (gfx1250) — compile-verified
//
#include <hip/hip_runtime.h>

typedef __attribute__((ext_vector_type(2))) float v2f;
typedef __attribute__((ext_vector_type(8))) float v8f;

#define S_TOK   1024
#define E_DIM   2048
#define H_HEADS 16
#define D_HEAD  128
#define L_CTX   2048
#define NPAST   1024
#define N3E     (3 * E_DIM)

// ---------------------------------------------------------------------------
// Kernel 1: QKV GEMM.  C[1024,6144] = x @ W1 + b1.
// One wave computes a 32(M) x 64(N) block: 2 A-frags x 4 B-frags -> 8 WMMAs
// per K-step of 4 (v_wmma_f32_16x16x4_f32), doubling B reuse vs a 16-row
// strip.  Q -> workspace [S,E]; K/V scattered into cache rows NPAST+s.
// ---------------------------------------------------------------------------
__global__ __launch_bounds__(32) void qkv_gemm_kernel(
    const float* __restrict__ x, const float* __restrict__ W1,
    const float* __restrict__ bias1,
    float* __restrict__ q_ws, float* __restrict__ kcache,
    float* __restrict__ vcache)
{
    const int lane = threadIdx.x;
    const int ln   = lane & 15;
    const int half = lane >> 4;
    const int m0   = blockIdx.y * 32;
    const int n0   = blockIdx.x * 64;

    v8f acc[2][4] = {};
    const float* arow0 = x + (size_t)(m0 + ln) * E_DIM;
    const float* arow1 = arow0 + (size_t)16 * E_DIM;

    for (int k0 = 0; k0 < E_DIM; k0 += 4) {
        v2f a0 = *(const v2f*)(arow0 + k0 + 2 * half);
        v2f a1 = *(const v2f*)(arow1 + k0 + 2 * half);
        const float* brow0 = W1 + (size_t)(k0 + 2 * half) * N3E + n0 + ln;
        const float* brow1 = brow0 + N3E;
#pragma unroll
        for (int t = 0; t < 4; ++t) {
            v2f b;
            b.x = brow0[t * 16];
            b.y = brow1[t * 16];
            acc[0][t] = __builtin_amdgcn_wmma_f32_16x16x4_f32(
                false, a0, false, b, (short)0, acc[0][t], false, false);
            acc[1][t] = __builtin_amdgcn_wmma_f32_16x16x4_f32(
                false, a1, false, b, (short)0, acc[1][t], false, false);
        }
    }

#pragma unroll
    for (int mi = 0; mi < 2; ++mi) {
#pragma unroll
        for (int t = 0; t < 4; ++t) {
            const int col = n0 + t * 16 + ln;
            const float bv = bias1[col];
#pragma unroll
            for (int r = 0; r < 8; ++r) {
                const int row = m0 + mi * 16 + r + 8 * half;
                const float v = acc[mi][t][r] + bv;
                if (col < E_DIM) {
                    q_ws[(size_t)row * E_DIM + col] = v;
                } else if (col < 2 * E_DIM) {
                    const int c = col - E_DIM;
                    kcache[((size_t)(c >> 7) * L_CTX + (NPAST + row)) * D_HEAD + (c & 127)] = v;
                } else {
                    const int c = col - 2 * E_DIM;
                    vcache[((size_t)(c >> 7) * L_CTX + (NPAST + row)) * D_HEAD + (c & 127)] = v;
                }
            }
        }
    }
}

// ---------------------------------------------------------------------------
// Kernel 2: flash attention, one wave per (head, 16-query tile).
// K/V tiles are double-buffered in LDS via the CDNA5 async global->LDS path
// (global_load_async_to_lds_b128 + s_wait_asynccnt): tile kt+1 streams into
// LDS while tile kt runs through the WMMA pipeline.  Q pre-scaled by
// 1/sqrt(D) in LDS; online softmax with width-16 shuffles; P staged through
// LDS for the C-layout -> A-layout transpose; WMMA for both q@k^T and p@v.
// ---------------------------------------------------------------------------
__global__ __launch_bounds__(32) void attn_kernel(
    const float* __restrict__ q_ws, const float* __restrict__ kcache,
    const float* __restrict__ vcache, float* __restrict__ a_ws)
{
    __shared__ float q_lds[16 * D_HEAD];
    __shared__ float p_lds[16 * 16];
    __shared__ float kvbuf[2][2][16 * D_HEAD];   // [buf][k/v][row*128+col]

    const int lane = threadIdx.x;
    const int ln   = lane & 15;
    const int half = lane >> 4;
    const int qt   = blockIdx.x;
    const int h    = blockIdx.y;
    const int q0   = qt * 16;
    const float scale = 0.08838834764831845f;    // 1/sqrt(128)

    // Stage the 16x128 Q tile into LDS, pre-scaled.
    for (int idx = lane; idx < 512; idx += 32) {
        const int row = idx >> 5;
        const int c4  = (idx & 31) * 4;
        const float* src = q_ws + (size_t)(q0 + row) * E_DIM + h * D_HEAD + c4;
        float4 v = *(const float4*)src;
        v.x *= scale; v.y *= scale; v.z *= scale; v.w *= scale;
        *(float4*)(&q_lds[row * D_HEAD + c4]) = v;
    }
    __syncthreads();

    const float* Kc = kcache + (size_t)h * L_CTX * D_HEAD;
    const float* Vc = vcache + (size_t)h * L_CTX * D_HEAD;

    // Wave-relative LDS byte offset of the K/V staging buffers (flat->LDS
    // keeps the offset in the low 32 bits).
    const unsigned kv_base = (unsigned)(uintptr_t)(&kvbuf[0][0][0]);

    // Async-prefetch one 16x128 K tile + 16x128 V tile into LDS buffer `buf`:
    // 32 x global_load_async_to_lds_b128 (one row per instruction, 16B/lane).
    auto prefetch = [&](int buf, int j0) {
        // DS and ASYNC ops are mutually unordered: make sure prior ds_loads
        // from this buffer have drained before the async engine rewrites it.
        asm volatile("s_wait_dscnt 0x0" ::: "memory");
        const unsigned kdst0 = kv_base + (unsigned)buf * (2u * 16u * D_HEAD * 4u)
                             + (unsigned)lane * 16u;
        const unsigned vdst0 = kdst0 + 16u * D_HEAD * 4u;
#pragma unroll
        for (int row = 0; row < 16; ++row) {
            const unsigned long long gk =
                (unsigned long long)(uintptr_t)(Kc + (size_t)(j0 + row) * D_HEAD + lane * 4);
            const unsigned long long gv =
                (unsigned long long)(uintptr_t)(Vc + (size_t)(j0 + row) * D_HEAD + lane * 4);
            const unsigned kd = kdst0 + (unsigned)row * 512u;
            const unsigned vd = vdst0 + (unsigned)row * 512u;
            asm volatile("global_load_async_to_lds_b128 %0, %1, off"
                         :: "v"(kd), "v"(gk) : "memory");
            asm volatile("global_load_async_to_lds_b128 %0, %1, off"
                         :: "v"(vd), "v"(gv) : "memory");
        }
    };

    float m_run[8], l_run[8];
    v8f o_acc[8] = {};
#pragma unroll
    for (int r = 0; r < 8; ++r) { m_run[r] = -__builtin_inff(); l_run[r] = 0.0f; }

    const int nkt = qt + (NPAST / 16) + 1;   // causal: tiles with any valid key
    prefetch(0, 0);

    for (int kt = 0; kt < nkt; ++kt) {
        const int j0  = kt * 16;
        const int cur = kt & 1;

        if (kt + 1 < nkt) {
            prefetch(cur ^ 1, j0 + 16);
            // Async loads complete in order: <=32 outstanding means the
            // current tile's 32 transfers have landed in LDS.
            asm volatile("s_wait_asynccnt 0x20" ::: "memory");
        } else {
            asm volatile("s_wait_asynccnt 0x0" ::: "memory");
        }
        const float* kb = &kvbuf[cur][0][0];
        const float* vb = &kvbuf[cur][1][0];

        // ---- scores: S = Q @ K^T  (K-dim = 128, 32 WMMA, all from LDS) ----
        v8f sacc = {};
#pragma unroll 4
        for (int d0 = 0; d0 < D_HEAD; d0 += 4) {
            v2f a = *(const v2f*)(&q_lds[ln * D_HEAD + d0 + 2 * half]);
            v2f b = *(const v2f*)(&kb[ln * D_HEAD + d0 + 2 * half]);
            sacc = __builtin_amdgcn_wmma_f32_16x16x4_f32(
                false, a, false, b, (short)0, sacc, false, false);
        }

        float s[8];
        const bool diag = (kt == nkt - 1);
#pragma unroll
        for (int r = 0; r < 8; ++r) {
            s[r] = sacc[r];
            if (diag && (q0 + r + 8 * half + NPAST < j0 + ln)) s[r] = -1e10f;
        }

        // ---- online softmax (reductions across the 16 lanes of each half) --
#pragma unroll
        for (int r = 0; r < 8; ++r) {
            float mx = s[r];
            mx = fmaxf(mx, __shfl_xor(mx, 1, 16));
            mx = fmaxf(mx, __shfl_xor(mx, 2, 16));
            mx = fmaxf(mx, __shfl_xor(mx, 4, 16));
            mx = fmaxf(mx, __shfl_xor(mx, 8, 16));
            const float mnew = fmaxf(m_run[r], mx);
            const float p    = __expf(s[r] - mnew);
            float rs = p;
            rs += __shfl_xor(rs, 1, 16);
            rs += __shfl_xor(rs, 2, 16);
            rs += __shfl_xor(rs, 4, 16);
            rs += __shfl_xor(rs, 8, 16);
            const float corr = __expf(m_run[r] - mnew);
            l_run[r] = l_run[r] * corr + rs;
            m_run[r] = mnew;
#pragma unroll
            for (int dt = 0; dt < 8; ++dt) o_acc[dt][r] *= corr;
            p_lds[(r + 8 * half) * 16 + ln] = p;   // C-layout -> LDS
        }
        __syncthreads();

        // ---- O += P @ V  (K-dim = 16 -> 4 chunks; 8 N-tiles over D=128) ----
        v2f af[4];
#pragma unroll
        for (int kk = 0; kk < 4; ++kk)
            af[kk] = *(const v2f*)(&p_lds[ln * 16 + kk * 4 + 2 * half]);
#pragma unroll
        for (int dt = 0; dt < 8; ++dt) {
#pragma unroll
            for (int kk = 0; kk < 4; ++kk) {
                const int krow = kk * 4 + 2 * half;
                v2f b;
                b.x = vb[krow * D_HEAD + dt * 16 + ln];
                b.y = vb[(krow + 1) * D_HEAD + dt * 16 + ln];
                o_acc[dt] = __builtin_amdgcn_wmma_f32_16x16x4_f32(
                    false, af[kk], false, b, (short)0, o_acc[dt], false, false);
            }
        }
        __syncthreads();   // p_lds reused next tile
    }

    // ---- normalize and write merged-head output a_ws[S, E] ----
#pragma unroll
    for (int dt = 0; dt < 8; ++dt) {
#pragma unroll
        for (int r = 0; r < 8; ++r) {
            const int row = q0 + r + 8 * half;
            a_ws[(size_t)row * E_DIM + h * D_HEAD + dt * 16 + ln] =
                o_acc[dt][r] / l_run[r];
        }
    }
}

// ---------------------------------------------------------------------------
// Kernel 3: output projection.  out[1024,2048] = a_ws @ W2 + b2.
// Same 32x64 per-wave WMMA blocking as kernel 1.
// ---------------------------------------------------------------------------
__global__ __launch_bounds__(32) void proj_gemm_kernel(
    const float* __restrict__ a_ws, const float* __restrict__ W2,
    const float* __restrict__ bias2, float* __restrict__ out)
{
    const int lane = threadIdx.x;
    const int ln   = lane & 15;
    const int half = lane >> 4;
    const int m0   = blockIdx.y * 32;
    const int n0   = blockIdx.x * 64;

    v8f acc[2][4] = {};
    const float* arow0 = a_ws + (size_t)(m0 + ln) * E_DIM;
    const float* arow1 = arow0 + (size_t)16 * E_DIM;

    for (int k0 = 0; k0 < E_DIM; k0 += 4) {
        v2f a0 = *(const v2f*)(arow0 + k0 + 2 * half);
        v2f a1 = *(const v2f*)(arow1 + k0 + 2 * half);
        const float* brow0 = W2 + (size_t)(k0 + 2 * half) * E_DIM + n0 + ln;
        const float* brow1 = brow0 + E_DIM;
#pragma unroll
        for (int t = 0; t < 4; ++t) {
            v2f b;
            b.x = brow0[t * 16];
            b.y = brow1[t * 16];
            acc[0][t] = __builtin_amdgcn_wmma_f32_16x16x4_f32(
                false, a0, false, b, (short)0, acc[0][t], false, false);
            acc[1][t] = __builtin_amdgcn_wmma_f32_16x16x4_f32(
                false, a1, false, b, (short)0, acc[1][t], false, false);
        }
    }

#pragma unroll
    for (int mi = 0; mi < 2; ++mi) {
#pragma unroll
        for (int t = 0; t < 4; ++t) {
            const int col = n0 + t * 16 + ln;
            const float bv = bias2[col];
#pragma unroll
            for (int r = 0; r < 8; ++r) {
                const int row = m0 + mi * 16 + r + 8 * half;
                out[(size_t)row * E_DIM + col] = acc[mi][t][r] + bv;
            }
        }
    }
}

// ---------------------------------------------------------------------------
extern "C" void kernel_launch(void* const* d_in, const int* in_sizes, int n_in,
                              void* d_out, int out_size, void* d_ws, size_t ws_size,
                              hipStream_t stream)
{
    (void)in_sizes; (void)n_in; (void)out_size; (void)ws_size;

    const float* x     = (const float*)d_in[0];
    const float* past  = (const float*)d_in[1];
    const float* W1    = (const float*)d_in[2];
    const float* b1    = (const float*)d_in[3];
    const float* W2    = (const float*)d_in[4];
    const float* b2    = (const float*)d_in[5];

    float* out    = (float*)d_out;                       // [S, E]
    float* cache  = out + (size_t)S_TOK * E_DIM;         // [2, H, L, D]
    float* kcache = cache;                               // [H, L, D]
    float* vcache = cache + (size_t)H_HEADS * L_CTX * D_HEAD;

    float* q_ws = (float*)d_ws;                          // [S, E]
    float* a_ws = q_ws + (size_t)S_TOK * E_DIM;          // [S, E]

    // 1. new_cache starts as a copy of past (rows NPAST.. overwritten below).
    hipMemcpyAsync(cache, past,
                   sizeof(float) * 2ull * H_HEADS * L_CTX * D_HEAD,
                   hipMemcpyDeviceToDevice, stream);

    // 2. QKV projection + fused KV-cache update.
    qkv_gemm_kernel<<<dim3(N3E / 64, S_TOK / 32), 32, 0, stream>>>(
        x, W1, b1, q_ws, kcache, vcache);

    // 3. Causal flash attention with async K/V double-buffering.
    attn_kernel<<<dim3(S_TOK / 16, H_HEADS), 32, 0, stream>>>(
        q_ws, kcache, vcache, a_ws);

    // 4. Output projection.
    proj_gemm_kernel<<<dim3(E_DIM / 64, S_TOK / 32), 32, 0, stream>>>(
        a_ws, W2, b2, out);
}